// Attention_31241592111555
// MI455X (gfx1250) — compile-verified
//
#include <hip/hip_runtime.h>

typedef __attribute__((ext_vector_type(2))) float v2f;
typedef __attribute__((ext_vector_type(8))) float v8f;
typedef __attribute__((ext_vector_type(4))) int   v4i;

#define DD 128   // hidden_channels
#define HH 64    // hidden_size
#define KK 32    // nodes per hyperedge
#define EE 8192  // num hyperedges
#define NB 2048  // persistent blocks
#define EPB (EE / NB)   // 4 edges per block

#define GLOBAL_AS __attribute__((address_space(1)))
#define LDS_AS    __attribute__((address_space(3)))

__device__ __forceinline__ void async_copy16(const float* g, float* l) {
    // memory -> LDS, 16 bytes per lane, tracked by ASYNCcnt (no VGPR data path)
    __builtin_amdgcn_global_load_async_to_lds_b128(
        (GLOBAL_AS v4i*)g, (LDS_AS v4i*)l, /*imm_offset=*/0, /*cpol=*/0);
}

__launch_bounds__(256)
__global__ void hyperedge_attn_kernel(const float* __restrict__ X,
                                      const int*   __restrict__ idx,
                                      const float* __restrict__ W1,
                                      const float* __restrict__ b1,
                                      const float* __restrict__ W2,
                                      const float* __restrict__ b2,
                                      float* __restrict__ Zout,
                                      float* __restrict__ betaOut)
{
    __shared__ float sX [2 * KK * DD];   // 2 x 16 KB gathered node features
    __shared__ float sW1[2 * HH * DD];   // 2 x 32 KB per-edge W1
    __shared__ float sH [KK * HH];       // 8 KB hidden activations
    __shared__ float sW2[2 * HH];
    __shared__ float sBeta[KK];

    const int tid  = threadIdx.x;   // 0..255, 8 wave32 waves
    const int lane = tid & 31;
    const int wave = tid >> 5;

    // Issue async staging of edge e into LDS buffer `buf` (12 B128 ops/thread).
    auto stage = [&](int e, int buf) {
        const int* eidx = idx + e * KK;
        float* bX = sX + buf * KK * DD;
        for (int i = tid; i < KK * (DD / 4); i += 256) {     // 4 per thread
            const int  k   = i >> 5;
            const int  c   = i & 31;
            const long row = (long)eidx[k];
            async_copy16(X + row * DD + c * 4, bX + k * DD + c * 4);
        }
        const float* w1e = W1 + (long)e * HH * DD;
        float* bW = sW1 + buf * HH * DD;
        for (int i = tid; i < HH * (DD / 4); i += 256)       // 8 per thread
            async_copy16(w1e + 4 * i, bW + 4 * i);
        if (tid < HH) sW2[buf * HH + tid] = W2[e * HH + tid];
    };

    int e = blockIdx.x;
    stage(e, 0);                                 // prologue fill

    for (int it = 0; it < EPB; ++it) {
        const int  buf  = it & 1;
        const bool more = (it + 1 < EPB);

        if (more) {
            stage(e + NB, buf ^ 1);              // prefetch next edge
            asm volatile("s_wait_asynccnt 12" ::: "memory");  // current edge done
        } else {
            asm volatile("s_wait_asynccnt 0" ::: "memory");
        }
        __syncthreads();                         // all waves' async data visible

        const float* bX = sX  + buf * KK * DD;
        const float* bW = sW1 + buf * HH * DD;

        // ---- GEMM1: h = leaky_relu(Xe @ W1^T + b1)  [32x64], 1 tile/wave ----
        {
            const int mt   = wave >> 2;
            const int nt   = wave & 3;
            const int half = lane >> 4;
            const int l    = lane & 15;
            const float* aRow = bX + (mt * 16 + l) * DD + 2 * half;
            const float* bRow = bW + (nt * 16 + l) * DD + 2 * half;

            v8f acc = {};
            #pragma unroll
            for (int k = 0; k < DD; k += 4) {
                v2f a = *(const v2f*)(aRow + k);
                v2f b = *(const v2f*)(bRow + k);
                acc = __builtin_amdgcn_wmma_f32_16x16x4_f32(
                    false, a, false, b, (short)0, acc, false, false);
            }

            const int   col = nt * 16 + l;
            const float bb  = b1[e * HH + col];
            #pragma unroll
            for (int r = 0; r < 8; ++r) {
                float h = acc[r] + bb;
                h = (h > 0.0f) ? h : 0.01f * h;            // leaky_relu 0.01
                const int row = mt * 16 + 8 * half + r;
                sH[row * HH + col] = h;
            }
        }
        __syncthreads();

        // ---- w = h @ W2 + b2 ; beta = softmax_k(w)  (wave 0) ----------------
        if (wave == 0) {
            float w = b2[e];
            const float* hr  = sH + lane * HH;
            const float* w2b = sW2 + buf * HH;
            #pragma unroll 8
            for (int h = 0; h < HH; ++h) w += hr[h] * w2b[h];

            float m = w;
            #pragma unroll
            for (int off = 16; off >= 1; off >>= 1)
                m = fmaxf(m, __shfl_xor(m, off, 32));
            const float p = __expf(w - m);
            float s = p;
            #pragma unroll
            for (int off = 16; off >= 1; off >>= 1)
                s += __shfl_xor(s, off, 32);
            const float beta = p / s;
            sBeta[lane] = beta;
            betaOut[e * KK + lane] = beta;
        }
        __syncthreads();

        // ---- Z = tanh(leaky_relu(beta^T @ Xe)) ------------------------------
        if (tid < DD) {
            float z = 0.0f;
            #pragma unroll 8
            for (int k = 0; k < KK; ++k) z += sBeta[k] * bX[k * DD + tid];
            z = (z > 0.0f) ? z : 0.01f * z;
            Zout[e * DD + tid] = tanhf(z);
        }
        __syncthreads();   // bX readers done before this buffer is re-staged
        e += NB;
    }
}

extern "C" void kernel_launch(void* const* d_in, const int* in_sizes, int n_in,
                              void* d_out, int out_size, void* d_ws, size_t ws_size,
                              hipStream_t stream) {
    const float* X   = (const float*)d_in[0];
    const int*   idx = (const int*)  d_in[1];
    const float* W1  = (const float*)d_in[2];
    const float* b1  = (const float*)d_in[3];
    const float* W2  = (const float*)d_in[4];
    const float* b2  = (const float*)d_in[5];

    float* Z    = (float*)d_out;           // [E, 128] first
    float* beta = Z + (size_t)EE * DD;     // then [E, 32]

    hipLaunchKernelGGL(hyperedge_attn_kernel, dim3(NB), dim3(256), 0, stream,
                       X, idx, W1, b1, W2, b2, Z, beta);
}